// RoI_17188459118745
// MI455X (gfx1250) — compile-verified
//
#include <hip/hip_runtime.h>
#include <hip/hip_bf16.h>
#include <math.h>

// Problem dims (fixed by setup_inputs)
#define BATCH 8
#define XDIM  128          // H
#define YDIM  128          // W
#define CDIM  128
#define NROI  128
#define PH    7
#define PW    7

#define NWAVES 8
#define CHUNK  8           // pixels staged per async batch
#define C4     (CDIM / 4)  // 32 float4 per pixel (one per lane)

__device__ __forceinline__ float4 f4max(float4 a, float4 b) {
    return make_float4(__builtin_fmaxf(a.x, b.x), __builtin_fmaxf(a.y, b.y),
                       __builtin_fmaxf(a.z, b.z), __builtin_fmaxf(a.w, b.w));
}

__global__ __launch_bounds__(NWAVES * 32) void roi_maxpool_async(
    const float* __restrict__ feat,   // [B, X, Y, C]
    const int*   __restrict__ rois,   // [B, N, 4] = (minX, minY, maxX, maxY)
    float*       __restrict__ out)    // [B, N, PH, PW, C]
{
    // Per-wave double-buffered staging: 8 waves * 2 bufs * 8 px * 512B = 64 KB
    __shared__ float4 lbuf[NWAVES][2][CHUNK * C4];

    const int bn = blockIdx.x;        // b*NROI + n
    const int b  = bn >> 7;

    const int4 roi  = ((const int4*)rois)[bn];
    const int  minX = roi.x, minY = roi.y, maxX = roi.z, maxY = roi.w;
    const int  dx   = (maxX - minX) / PW;
    const int  dy   = (maxY - minY) / PH;

    const int wave = threadIdx.x >> 5;
    const int lane = threadIdx.x & 31;

    // Byte offset of this image + this lane's channel quad (fits in u32: < 64 MB)
    const unsigned imgBase =
        (unsigned)b * (unsigned)(XDIM * YDIM * CDIM * 4) + (unsigned)lane * 16u;

    float4* __restrict__ out4 =
        (float4*)out + (size_t)bn * (PH * PW * C4) + lane;

    for (int bin = wave; bin < PH * PW; bin += NWAVES) {   // uniform per wave
        const int h  = bin / PW;                // y-bin
        const int w  = bin - h * PW;            // x-bin
        const int xs = minX + w * dx;
        const int xe = (w == PW - 1) ? maxX : (minX + (w + 1) * dx);
        const int ys = minY + h * dy;
        const int ye = (h == PH - 1) ? maxY : (minY + (h + 1) * dy);

        const int nPix    = (xe - xs) * (ye - ys);   // >= 4 for valid ROIs
        const int nChunks = (nPix + CHUNK - 1) / CHUNK;

        int cx = xs, cy = ys;   // pixel cursor (advanced only by issue, sticks at last px)

        auto issue = [&](int cbuf) {
            const unsigned lds0 =
                (unsigned)(size_t)(&lbuf[wave][cbuf][0]);   // low 32b = LDS offset
            // WAR guard: prior ds_loads from this buffer must have retired
            asm volatile("s_wait_dscnt 0x0" ::: "memory");
#pragma unroll
            for (int i = 0; i < CHUNK; ++i) {
                const unsigned goff =
                    imgBase + (unsigned)cx * (unsigned)(YDIM * CDIM * 4)
                            + (unsigned)cy * (unsigned)(CDIM * 4);
                const unsigned laddr = lds0 + (unsigned)(i * CDIM * 4) + (unsigned)lane * 16u;
                // CDNA5 async copy: 32 lanes x 16B = one pixel's 512B channel row -> LDS
                asm volatile("global_load_async_to_lds_b128 %0, %1, %2"
                             :: "v"(laddr), "v"(goff), "s"(feat)
                             : "memory");
                // advance; pad by replaying last pixel (idempotent under max)
                if (++cy == ye) {
                    if (cx + 1 < xe) { ++cx; cy = ys; }
                    else             { cy = ye - 1;   }
                }
            }
        };

        issue(0);
        float4 m0 = make_float4(-INFINITY, -INFINITY, -INFINITY, -INFINITY);
        float4 m1 = m0;

        for (int c = 0; c < nChunks; ++c) {
            const int cbuf = c & 1;
            if (c + 1 < nChunks) {
                issue((c + 1) & 1);
                // <=8 outstanding: async loads complete in order, so chunk c landed
                asm volatile("s_wait_asynccnt 0x8" ::: "memory");
            } else {
                asm volatile("s_wait_asynccnt 0x0" ::: "memory");
            }
#pragma unroll
            for (int i = 0; i < CHUNK; i += 2) {
                m0 = f4max(m0, lbuf[wave][cbuf][(i + 0) * C4 + lane]);
                m1 = f4max(m1, lbuf[wave][cbuf][(i + 1) * C4 + lane]);
            }
        }

        out4[bin * C4] = f4max(m0, m1);
    }
}

extern "C" void kernel_launch(void* const* d_in, const int* in_sizes, int n_in,
                              void* d_out, int out_size, void* d_ws, size_t ws_size,
                              hipStream_t stream) {
    (void)in_sizes; (void)n_in; (void)d_ws; (void)ws_size; (void)out_size;
    const float* feat = (const float*)d_in[0];
    const int*   rois = (const int*)d_in[1];
    float*       out  = (float*)d_out;

    dim3 grid(BATCH * NROI);     // one block per ROI
    dim3 block(NWAVES * 32);     // 8 wave32 waves
    roi_maxpool_async<<<grid, block, 0, stream>>>(feat, rois, out);
}